// RollingStepKVCache_58772332479038
// MI455X (gfx1250) — compile-verified
//
#include <hip/hip_runtime.h>
#include <cstdint>

// Shapes fixed by setup_inputs(): B=16, H=8, L=4096, D=128, S=16 (S read from device).
constexpr int  Bq   = 16;
constexpr int  Hq   = 8;
constexpr int  Lq   = 4096;                        // power of two (log2 = 12)
constexpr int  Dq   = 128;
constexpr long N4   = (long)Bq * Hq * Lq * (Dq/4); // 16,777,216 float4 per tensor
constexpr long TOTAL4 = 2 * N4;                    // k then v concatenated in d_out
constexpr int  CHUNK  = 16;                        // float4 per chunk (256 B, half a row)
constexpr long NC     = N4 / CHUNK;                // chunks per tensor
constexpr long NCHUNK = TOTAL4 / CHUNK;            // 2,097,152 total chunks
constexpr int  TPB    = 256;

__global__ __launch_bounds__(TPB)
void rolling_kv_async_copy(const float* __restrict__ kc,
                           const float* __restrict__ vc,
                           const float* __restrict__ kv,
                           const float* __restrict__ vv,
                           const int*   __restrict__ stepp,
                           float* __restrict__ out)
{
    // Per-thread LDS staging slice: CHUNK*16 = 256 B -> 64 KB per 256-thread block.
    __shared__ __align__(16) unsigned char lds[TPB * CHUNK * 16];

    const int  S    = *stepp;                      // wave-uniform (== 16)
    const long keep = (long)Lq - S;                // rows kept from the old cache
    const long gt   = (long)blockIdx.x * TPB + threadIdx.x;
    const long gstr = (long)gridDim.x * TPB;
    // Low 32 bits of the generic shared address are the LDS byte offset.
    const uint32_t ldsBase =
        (uint32_t)(uintptr_t)&lds[(size_t)threadIdx.x * (CHUNK * 16)];

    for (long c = gt; c < NCHUNK; c += gstr) {
        // ---- one-time per-chunk address math (contiguous 256 B chunk) ----
        long i = c;
        const float* cache = kc;
        const float* val   = kv;
        if (i >= NC) { i -= NC; cache = vc; val = vv; }   // second tensor = V
        long row  = i >> 1;            // global row index: (b*H + h)*L + l
        long half = i & 1;             // which 256 B half of the 512 B row
        long l    = row & (Lq - 1);    // sequence position
        long bh   = row >> 12;         // (b*H + h)
        uint64_t src;
        if (l < keep)
            src = (uint64_t)cache + (uint64_t)(row + S) * 512u + (uint64_t)half * 256u;
        else
            src = (uint64_t)val + (uint64_t)(bh * S + (l - keep)) * 512u
                                + (uint64_t)half * 256u;
        uint64_t dst = (uint64_t)out + (uint64_t)c * 256u;

        // ---- 16 async 128b loads HBM -> LDS; IOFFSET advances both sides ----
#pragma unroll
        for (int j = 0; j < CHUNK; ++j)
            asm volatile("global_load_async_to_lds_b128 %0, %1, off offset:%c2"
                         :: "v"(ldsBase), "v"(src), "n"(j * 16) : "memory");
        // All loads done -> LDS slice valid (ASYNCcnt drained).
        asm volatile("s_wait_asynccnt 0" ::: "memory");

        // ---- 16 async 128b stores LDS -> HBM ----
#pragma unroll
        for (int j = 0; j < CHUNK; ++j)
            asm volatile("global_store_async_from_lds_b128 %0, %1, off offset:%c2"
                         :: "v"(dst), "v"(ldsBase), "n"(j * 16) : "memory");
        // Drain stores before re-using the LDS slice next iteration.
        asm volatile("s_wait_asynccnt 0" ::: "memory");
    }
}

extern "C" void kernel_launch(void* const* d_in, const int* in_sizes, int n_in,
                              void* d_out, int out_size, void* d_ws, size_t ws_size,
                              hipStream_t stream)
{
    const float* kc   = (const float*)d_in[0];  // k_cache (B,H,L,D)
    const float* vc   = (const float*)d_in[1];  // v_cache (B,H,L,D)
    const float* kv   = (const float*)d_in[2];  // k_val   (B,H,S,D)
    const float* vv   = (const float*)d_in[3];  // v_val   (B,H,S,D)
    const int*   step = (const int*)d_in[4];    // step_size scalar on device
    float*       out  = (float*)d_out;          // [k_new | v_new] flat

    dim3 grid(4096), block(TPB);
    hipLaunchKernelGGL(rolling_kv_async_copy, grid, block, 0, stream,
                       kc, vc, kv, vv, step, out);
}